// Lig_Update_5841155522617
// MI455X (gfx1250) — compile-verified
//
#include <hip/hip_runtime.h>
#include <math.h>

typedef __attribute__((ext_vector_type(2))) float v2f;
typedef __attribute__((ext_vector_type(8))) float v8f;

#define IN_DIM 128
#define GG 4096      // num_graphs (reference constant; device scalar can't size grids)
#define WPB 4        // waves per block in the va-pass kernels (wave32)

// ---- ordered-uint encoding for float atomic max ----
__device__ __forceinline__ unsigned fenc(float x) {
  unsigned u = __float_as_uint(x);
  return (u & 0x80000000u) ? ~u : (u | 0x80000000u);
}
__device__ __forceinline__ float fdec(unsigned e) {
  return (e & 0x80000000u) ? __uint_as_float(e ^ 0x80000000u)
                           : __uint_as_float(~e);
}

// K0: zero scratch
__global__ void k0_zero(float* __restrict__ p, int n) {
  int i = blockIdx.x * blockDim.x + threadIdx.x;
  int stride = gridDim.x * blockDim.x;
  for (; i < n; i += stride) p[i] = 0.f;
}

// K1: h = va @ WB + bB  via V_WMMA_F32_16X16X4_F32 (N padded 8->16),
//     store h to scratch, atomic-max into per-graph m (ordered-uint).
__global__ __launch_bounds__(WPB * 32) void k1_h(
    const float* __restrict__ va, const float* __restrict__ WB,
    const float* __restrict__ bB, const int* __restrict__ batch_a,
    float* __restrict__ h_e, unsigned* __restrict__ m_ord, int Na) {
  __shared__ float lva[WPB][16 * IN_DIM];
  const int lane = threadIdx.x & 31;
  const int wid  = threadIdx.x >> 5;
  const int r0   = (blockIdx.x * WPB + wid) * 16;
  if (r0 >= Na) return;

  float* tile = lva[wid];
  #pragma unroll
  for (int j = 0; j < 16; ++j) {            // coalesced float4 staging
    int row = r0 + j; if (row >= Na) row = Na - 1;
    const float4 v = *(const float4*)(va + (size_t)row * IN_DIM + lane * 4);
    *(float4*)(tile + j * IN_DIM + lane * 4) = v;
  }
  asm volatile("s_wait_dscnt 0" ::: "memory");  // wave-local LDS RAW fence

  const int n  = lane & 15;          // A: M index ; B: N index
  const int kh = (lane >> 4) * 2;    // K sub-offset per lane-half
  const float bm = (n < 8) ? 1.f : 0.f;  // zero-pad B columns 8..15 (branchless)
  const int nc = n & 7;
  const float* arow = tile + n * IN_DIM;

  v8f acc = {0.f, 0.f, 0.f, 0.f, 0.f, 0.f, 0.f, 0.f};
  #pragma unroll 4
  for (int k0 = 0; k0 < IN_DIM; k0 += 4) {
    v2f a, b;
    a.x = arow[k0 + kh];
    a.y = arow[k0 + kh + 1];
    b.x = WB[(k0 + kh) * 8 + nc] * bm;
    b.y = WB[(k0 + kh + 1) * 8 + nc] * bm;
    acc = __builtin_amdgcn_wmma_f32_16x16x4_f32(false, a, false, b,
                                                (short)0, acc, false, false);
  }
  if (n < 8) {
    const float bb = bB[n];
    #pragma unroll
    for (int j = 0; j < 8; ++j) {
      const int M = j + 8 * (lane >> 4);     // D layout: vgpr j, lane half
      const int row = r0 + M;
      if (row < Na) {
        const float hv = acc[j] + bb;
        h_e[(size_t)row * 8 + n] = hv;
        const int g = batch_a[row];
        atomicMax(&m_ord[g * 8 + n], fenc(hv));
      }
    }
  }
}

// K2: e = exp(h - m[g]); accumulate per-graph denom
__global__ void k2_exp(float* __restrict__ h_e, const int* __restrict__ batch_a,
                       const unsigned* __restrict__ m_ord,
                       float* __restrict__ denom, int ne) {
  int i = blockIdx.x * blockDim.x + threadIdx.x;
  int stride = gridDim.x * blockDim.x;
  for (; i < ne; i += stride) {
    const int row = i >> 3, c = i & 7;
    const int g = batch_a[row];
    const float mv = fdec(m_ord[g * 8 + c]);
    const float e = __expf(h_e[i] - mv);
    h_e[i] = e;
    atomicAdd(&denom[g * 8 + c], e);
  }
}

// K3: s = va @ WA + bA via WMMA, then segmented outer-product accumulation
//     glo[g] += s_n (x) hh_n  exploiting sorted batch_a (few atomics/tile)
__global__ __launch_bounds__(WPB * 32) void k3_s_outer(
    const float* __restrict__ va, const float* __restrict__ WA,
    const float* __restrict__ bA, const int* __restrict__ batch_a,
    const float* __restrict__ h_e /* holds e */,
    const float* __restrict__ denom, float* __restrict__ glo, int Na) {
  __shared__ float lva[WPB][16 * IN_DIM];
  __shared__ float ls[WPB][16 * 64];
  __shared__ float lhh[WPB][16 * 8];
  __shared__ int   lg[WPB][16];
  const int lane = threadIdx.x & 31;
  const int wid  = threadIdx.x >> 5;
  const int r0   = (blockIdx.x * WPB + wid) * 16;
  if (r0 >= Na) return;

  float* tile = lva[wid];
  #pragma unroll
  for (int j = 0; j < 16; ++j) {
    int row = r0 + j; if (row >= Na) row = Na - 1;
    const float4 v = *(const float4*)(va + (size_t)row * IN_DIM + lane * 4);
    *(float4*)(tile + j * IN_DIM + lane * 4) = v;
  }
  if (lane < 16) {                       // stage hh = e/denom and graph ids
    int row = r0 + lane; if (row >= Na) row = Na - 1;
    const int g = batch_a[row];
    lg[wid][lane] = g;
    #pragma unroll
    for (int c = 0; c < 8; ++c)
      lhh[wid][lane * 8 + c] = h_e[(size_t)row * 8 + c] / denom[g * 8 + c];
  }
  asm volatile("s_wait_dscnt 0" ::: "memory");

  const int n  = lane & 15;
  const int kh = (lane >> 4) * 2;
  const float* arow = tile + n * IN_DIM;
  #pragma unroll
  for (int nt = 0; nt < 4; ++nt) {       // 16x64 output = 4 WMMA N-tiles
    v8f acc = {0.f, 0.f, 0.f, 0.f, 0.f, 0.f, 0.f, 0.f};
    #pragma unroll 4
    for (int k0 = 0; k0 < IN_DIM; k0 += 4) {
      v2f a, b;
      a.x = arow[k0 + kh];
      a.y = arow[k0 + kh + 1];
      b.x = WA[(k0 + kh) * 64 + nt * 16 + n];
      b.y = WA[(k0 + kh + 1) * 64 + nt * 16 + n];
      acc = __builtin_amdgcn_wmma_f32_16x16x4_f32(false, a, false, b,
                                                  (short)0, acc, false, false);
    }
    const float ba = bA[nt * 16 + n];
    #pragma unroll
    for (int j = 0; j < 8; ++j) {
      const int M = j + 8 * (lane >> 4);
      ls[wid][M * 64 + nt * 16 + n] = acc[j] + ba;
    }
  }
  asm volatile("s_wait_dscnt 0" ::: "memory");

  // each lane owns cols {lane, lane+32} x 8 softmax cols = 16 accumulators
  const int cA = lane, cB = lane + 32;
  float a0[8], a1[8];
  #pragma unroll
  for (int c = 0; c < 8; ++c) { a0[c] = 0.f; a1[c] = 0.f; }
  int curg = lg[wid][0];
  for (int m = 0; m < 16; ++m) {
    if (r0 + m >= Na) break;             // wave-uniform tail guard
    const int g = lg[wid][m];
    if (g != curg) {                     // segment boundary: flush to glo
      #pragma unroll
      for (int c = 0; c < 8; ++c) {
        atomicAdd(&glo[(size_t)curg * 512 + cA * 8 + c], a0[c]);
        atomicAdd(&glo[(size_t)curg * 512 + cB * 8 + c], a1[c]);
        a0[c] = 0.f; a1[c] = 0.f;
      }
      curg = g;
    }
    const float sA = ls[wid][m * 64 + cA];
    const float sB = ls[wid][m * 64 + cB];
    #pragma unroll
    for (int c = 0; c < 8; ++c) {
      const float h = lhh[wid][m * 8 + c];
      a0[c] += sA * h;
      a1[c] += sB * h;
    }
  }
  #pragma unroll
  for (int c = 0; c < 8; ++c) {
    atomicAdd(&glo[(size_t)curg * 512 + cA * 8 + c], a0[c]);
    atomicAdd(&glo[(size_t)curg * 512 + cB * 8 + c], a1[c]);
  }
}

// K4: per-graph: mean over 8 -> @WC + bC -> @WD + bD
__global__ __launch_bounds__(128) void k4_graph(
    const float* __restrict__ glo, const float* __restrict__ WC,
    const float* __restrict__ bC, const float* __restrict__ WD,
    const float* __restrict__ bD, float* __restrict__ glo2) {
  __shared__ float gm[64];
  __shared__ float gv[IN_DIM];
  const int g = blockIdx.x, t = threadIdx.x;
  if (t < 64) {
    float s = 0.f;
    #pragma unroll
    for (int c = 0; c < 8; ++c) s += glo[(size_t)g * 512 + t * 8 + c];
    gm[t] = s * 0.125f;
  }
  __syncthreads();
  {
    float s = bC[t];
    for (int k = 0; k < 64; ++k) s += gm[k] * WC[k * IN_DIM + t];
    gv[t] = s;
  }
  __syncthreads();
  if (t < 120) {
    float s = bD[t];
    for (int k = 0; k < IN_DIM; ++k) s += gv[k] * WD[k * 120 + t];
    glo2[(size_t)g * 120 + t] = s;
  }
}

// K5: per-graph node counts for batch_b
__global__ void k5_counts(const int* __restrict__ batch_b,
                          float* __restrict__ counts, int Nb) {
  int i = blockIdx.x * blockDim.x + threadIdx.x;
  int stride = gridDim.x * blockDim.x;
  for (; i < Nb; i += stride) atomicAdd(&counts[batch_b[i]], 1.0f);
}

// K6: total = counts @ glo2 ; w = WE @ total ; cb = bE . total
__global__ __launch_bounds__(128) void k6_total_w(
    const float* __restrict__ counts, const float* __restrict__ glo2,
    const float* __restrict__ WE, const float* __restrict__ bE,
    float* __restrict__ wvec /* [128]=w, [128]=cb */) {
  __shared__ float tl[120];
  const int t = threadIdx.x;
  if (t < 120) {
    float s = 0.f;
    for (int g = 0; g < GG; ++g) s += counts[g] * glo2[(size_t)g * 120 + t];
    tl[t] = s;
  }
  __syncthreads();
  float s = 0.f;
  for (int c = 0; c < 120; ++c) s += WE[t * 120 + c] * tl[c];
  wvec[t] = s;
  if (t == 0) {
    float cb = 0.f;
    for (int c = 0; c < 120; ++c) cb += bE[c] * tl[c];
    wvec[128] = cb;
  }
}

// K7: out = vb * (1 + sigmoid(vb . w + cb)) — one wave per row, fused
__global__ __launch_bounds__(256) void k7_out(
    const float* __restrict__ vb, const float* __restrict__ wvec,
    float* __restrict__ out, int Nb) {
  const int lane = threadIdx.x & 31;
  const int wid  = threadIdx.x >> 5;
  const int row  = blockIdx.x * 8 + wid;
  if (row >= Nb) return;
  const float4 v  = *(const float4*)(vb + (size_t)row * IN_DIM + lane * 4);
  const float4 wv = *(const float4*)(wvec + lane * 4);
  float p = v.x * wv.x + v.y * wv.y + v.z * wv.z + v.w * wv.w;
  p += __shfl_xor(p, 16);
  p += __shfl_xor(p, 8);
  p += __shfl_xor(p, 4);
  p += __shfl_xor(p, 2);
  p += __shfl_xor(p, 1);
  const float x = p + wvec[128];
  const float r = 1.f / (1.f + __expf(-x));
  const float sc = 1.f + r;
  float4 o;
  o.x = v.x * sc; o.y = v.y * sc; o.z = v.z * sc; o.w = v.w * sc;
  *(float4*)(out + (size_t)row * IN_DIM + lane * 4) = o;
}

extern "C" void kernel_launch(void* const* d_in, const int* in_sizes, int n_in,
                              void* d_out, int out_size, void* d_ws, size_t ws_size,
                              hipStream_t stream) {
  (void)n_in; (void)out_size; (void)ws_size;
  const float* va = (const float*)d_in[0];
  const float* vb = (const float*)d_in[1];
  const float* WA = (const float*)d_in[2];
  const float* bA = (const float*)d_in[3];
  const float* WB = (const float*)d_in[4];
  const float* bB = (const float*)d_in[5];
  const float* WC = (const float*)d_in[6];
  const float* bC = (const float*)d_in[7];
  const float* WD = (const float*)d_in[8];
  const float* bD = (const float*)d_in[9];
  const float* WE = (const float*)d_in[10];
  const float* bE = (const float*)d_in[11];
  const int* batch_a = (const int*)d_in[12];
  const int* batch_b = (const int*)d_in[13];
  const int Na = in_sizes[0] / IN_DIM;
  const int Nb = in_sizes[1] / IN_DIM;
  float* out = (float*)d_out;

  // workspace layout (~23.3 MB)
  char* p = (char*)d_ws;
  unsigned* m_ord = (unsigned*)p; p += (size_t)GG * 8 * 4;
  float* denom  = (float*)p;      p += (size_t)GG * 8 * 4;
  float* glo    = (float*)p;      p += (size_t)GG * 512 * 4;
  float* counts = (float*)p;      p += (size_t)GG * 4;
  float* glo2   = (float*)p;      p += (size_t)GG * 120 * 4;
  float* wvec   = (float*)p;      p += 256 * 4;
  float* h_e    = (float*)p;      // Na*8 floats

  const int nzero = GG * 8 + GG * 8 + GG * 512 + GG;  // m..counts contiguous
  k0_zero<<<512, 256, 0, stream>>>((float*)m_ord, nzero);

  const int tile_blocks = (Na + WPB * 16 - 1) / (WPB * 16);
  k1_h<<<tile_blocks, WPB * 32, 0, stream>>>(va, WB, bB, batch_a, h_e, m_ord, Na);

  const int ne = Na * 8;
  k2_exp<<<(ne + 255) / 256, 256, 0, stream>>>(h_e, batch_a, m_ord, denom, ne);

  k3_s_outer<<<tile_blocks, WPB * 32, 0, stream>>>(va, WA, bA, batch_a, h_e,
                                                   denom, glo, Na);
  k4_graph<<<GG, 128, 0, stream>>>(glo, WC, bC, WD, bD, glo2);
  k5_counts<<<512, 256, 0, stream>>>(batch_b, counts, Nb);
  k6_total_w<<<1, 128, 0, stream>>>(counts, glo2, WE, bE, wvec);
  k7_out<<<(Nb + 7) / 8, 256, 0, stream>>>(vb, wvec, out, Nb);
}